// Uni3FC_DINO_proj_68719477139
// MI455X (gfx1250) — compile-verified
//
#include <hip/hip_runtime.h>
#include <hip/hip_bf16.h>
#include <math.h>

#define IMG    224
#define SPIX   (IMG*IMG)
#define NPTS   16384
#define LRW    56
#define LRPIX  (LRW*LRW)
#define CVIEW  384
#define CTOT   1152

// workspace layout (floats)
#define WSUM_OFF   0        // 384: wsum[c] = sum_o W_up[c,o]
#define WNORM_OFF  384      // 1  : ||wsum||_2
#define PAR_OFF    392      // 12 * 8: {min_u, min_v, gs, off_u, off_v}
#define LR_OFF     488      // 12 * 3136: pooled 56x56 images

typedef float v2f __attribute__((ext_vector_type(2)));
typedef float v8f __attribute__((ext_vector_type(8)));

// rotate by -pi/2 around z (exact fp32 constants of np.cos/sin) and apply
// the per-view channel permutation; returns projected (u,v) and the z-value.
__device__ __forceinline__ void rot_view(const float* __restrict__ x, int b, int n, int view,
                                         float& u, float& v, float& z) {
  const float cc = 6.123233995736766e-17f;  // cos(-pi/2)
  const float ss = -1.0f;                   // sin(-pi/2)
  float X = x[(b*3+0)*NPTS + n];
  float Y = x[(b*3+1)*NPTS + n];
  float Z = x[(b*3+2)*NPTS + n];
  float p0 = cc*X + ss*Y;
  float p1 = -ss*X + cc*Y;
  float p2 = Z;
  if (view == 0)      { u = p0; v = p1; z = p2; }
  else if (view == 1) { u = p2; v = p0; z = p1; }
  else                { u = p1; v = p2; z = p0; }
}

// ---------------- kernel 0: wsum + ||wsum|| ----------------
__global__ void prep_w_kernel(const float* __restrict__ Wup, float* __restrict__ ws) {
  __shared__ float red[512];
  int tid = threadIdx.x;
  float sq = 0.f;
  if (tid < CVIEW) {
    float s = Wup[tid*3+0] + Wup[tid*3+1] + Wup[tid*3+2];
    ws[WSUM_OFF + tid] = s;
    sq = s * s;
  }
  red[tid] = sq;
  __syncthreads();
  for (int st = 256; st > 0; st >>= 1) {
    if (tid < st) red[tid] += red[tid + st];
    __syncthreads();
  }
  if (tid == 0) ws[WNORM_OFF] = sqrtf(red[0]);
}

// ---------------- kernel 1: per (view,batch) grid params ----------------
__global__ void params_kernel(const float* __restrict__ x, float* __restrict__ ws) {
  int imgid = blockIdx.x;            // 0..11
  int view = imgid >> 2, b = imgid & 3;
  int tid = threadIdx.x;
  float mnu = 3.4e38f, mxu = -3.4e38f, mnv = 3.4e38f, mxv = -3.4e38f;
  for (int n = tid; n < NPTS; n += blockDim.x) {
    float u, v, z; rot_view(x, b, n, view, u, v, z);
    mnu = fminf(mnu, u); mxu = fmaxf(mxu, u);
    mnv = fminf(mnv, v); mxv = fmaxf(mxv, v);
  }
  __shared__ float s0[256], s1[256], s2[256], s3[256];
  s0[tid]=mnu; s1[tid]=mxu; s2[tid]=mnv; s3[tid]=mxv;
  __syncthreads();
  for (int st = 128; st > 0; st >>= 1) {
    if (tid < st) {
      s0[tid] = fminf(s0[tid], s0[tid+st]);
      s1[tid] = fmaxf(s1[tid], s1[tid+st]);
      s2[tid] = fminf(s2[tid], s2[tid+st]);
      s3[tid] = fmaxf(s3[tid], s3[tid+st]);
    }
    __syncthreads();
  }
  if (tid == 0) {
    float minu = s0[0], maxu = s1[0], minv = s2[0], maxv = s3[0];
    float gs = fmaxf(maxu - minu, maxv - minv) / (float)(IMG - 3);
    // idx_min per dim is exactly 0, so dense.min = -1, dense.max = idxmax+3
    float idxmaxu = floorf((maxu - minu) / gs);
    float idxmaxv = floorf((maxv - minv) / gs);
    float cenu = floorf((idxmaxu + 2.0f) * 0.5f);
    float cenv = floorf((idxmaxv + 2.0f) * 0.5f);
    float* p = ws + PAR_OFF + imgid*8;
    p[0] = minu; p[1] = minv; p[2] = gs;
    p[3] = 111.0f - cenu;   // off_u = IMG/2 - center - 1
    p[4] = 111.0f - cenv;
  }
}

// ---------------- kernel 2: scatter into LDS image, normalize, 4x4 pool ----------------
// One workgroup per (view,batch). 224*224 f32 image (200 KB) lives entirely in
// CDNA5's 320 KB WGP LDS; z-scatter uses ds_add_f32 atomics.
__global__ void scatter_pool_kernel(const float* __restrict__ x, float* __restrict__ ws) {
  extern __shared__ float smem[];     // SPIX image + 32 reduction slots
  int imgid = blockIdx.x;
  int view = imgid >> 2, b = imgid & 3;
  int tid = threadIdx.x;
  int nthr = blockDim.x;              // 512
  const float* p = ws + PAR_OFF + imgid*8;
  float minu = p[0], minv = p[1], gs = p[2], offu = p[3], offv = p[4];
  float inv_gs = 1.0f / gs;

  for (int i = tid; i < SPIX; i += nthr) smem[i] = 0.f;
  __syncthreads();

  for (int n = tid; n < NPTS; n += nthr) {
    float u, v, z; rot_view(x, b, n, view, u, v, z);
    float iu = floorf((u - minu) * inv_gs);
    float iv = floorf((v - minv) * inv_gs);
    #pragma unroll
    for (int dx = -2; dx <= 2; ++dx)
      #pragma unroll
      for (int dy = -2; dy <= 2; ++dy) {
        float qx = iu + (float)dx + 1.0f + offu;
        float qy = iv + (float)dy + 1.0f + offv;
        qx += (qx < 0.f) ? 1.f : 0.f;  qx -= (qx > (float)(IMG-1)) ? 1.f : 0.f;
        qy += (qy < 0.f) ? 1.f : 0.f;  qy -= (qy > (float)(IMG-1)) ? 1.f : 0.f;
        int xx = min(max((int)qx, 0), IMG-1);
        int yy = min(max((int)qy, 0), IMG-1);
        atomicAdd(&smem[xx*IMG + yy], z);   // ds_add_f32
      }
  }
  __syncthreads();

  // min/max of d = (sigmoid(s)-mean)/std over all pixels
  float dmn = 3.4e38f, dmx = -3.4e38f;
  for (int i = tid; i < SPIX; i += nthr) {
    float s = smem[i];
    float d = (1.0f/(1.0f + expf(-s)) - 0.485f) * (1.0f/0.229f);
    dmn = fminf(dmn, d); dmx = fmaxf(dmx, d);
  }
  for (int m = 16; m > 0; m >>= 1) {            // wave32 butterfly
    dmn = fminf(dmn, __shfl_xor(dmn, m, 32));
    dmx = fmaxf(dmx, __shfl_xor(dmx, m, 32));
  }
  float* red = smem + SPIX;
  int wave = tid >> 5, lane = tid & 31, nw = nthr >> 5;
  if (lane == 0) { red[wave] = dmn; red[16 + wave] = dmx; }
  __syncthreads();
  if (tid == 0) {
    float a = red[0], c = red[16];
    for (int w = 1; w < nw; ++w) { a = fminf(a, red[w]); c = fmaxf(c, red[16 + w]); }
    red[30] = a; red[31] = c;
  }
  __syncthreads();
  float dmin = red[30], dmax = red[31];
  float inv_rng = 1.0f / (dmax - dmin);

  // img_out + 4x4 average pool -> lr (56x56), single read of each hi-res pixel
  float* lr = ws + LR_OFF + imgid*LRPIX;
  for (int pidx = tid; pidx < LRPIX; pidx += nthr) {
    int h = pidx / LRW, w = pidx % LRW;
    float acc = 0.f;
    #pragma unroll
    for (int i = 0; i < 4; ++i)
      #pragma unroll
      for (int j = 0; j < 4; ++j) {
        float s = smem[(h*4 + i)*IMG + (w*4 + j)];
        float d = (1.0f/(1.0f + expf(-s)) - 0.485f) * (1.0f/0.229f);
        float val = (s == 0.f) ? -1.f : (d - dmin) * inv_rng;
        acc += val;
      }
    lr[pidx] = acc * (1.0f/16.0f);
  }
}

// ---------------- kernel 3: bicubic gather + rank-1 WMMA expansion ----------------
__device__ __forceinline__ void cubic_w(float fr, float* w) {   // Keys a=-0.5
  float s;
  s = 1.0f + fr; w[0] = ((-0.5f*s + 2.5f)*s - 4.0f)*s + 2.0f;
  s = fr;        w[1] = (1.5f*s - 2.5f)*s*s + 1.0f;
  s = 1.0f - fr; w[2] = (1.5f*s - 2.5f)*s*s + 1.0f;
  s = 2.0f - fr; w[3] = ((-0.5f*s + 2.5f)*s - 4.0f)*s + 2.0f;
}

__global__ void __launch_bounds__(256)
expand_kernel(const float* __restrict__ x, const float* __restrict__ ws,
              float* __restrict__ out) {
  __shared__ float lr_s[LRPIX];
  __shared__ float wsum_s[CVIEW];
  __shared__ float par_s[8];
  int imgid = blockIdx.y;
  int view = imgid >> 2, b = imgid & 3;
  int tid = threadIdx.x;
  const float* lr_g = ws + LR_OFF + imgid*LRPIX;
  for (int i = tid; i < LRPIX; i += 256) lr_s[i] = lr_g[i];
  for (int i = tid; i < CVIEW; i += 256) wsum_s[i] = ws[WSUM_OFF + i];
  if (tid < 5) par_s[tid] = ws[PAR_OFF + imgid*8 + tid];
  if (tid == 5) par_s[5] = ws[WNORM_OFF];
  __syncthreads();
  float minu = par_s[0], minv = par_s[1], gs = par_s[2];
  float offu = par_s[3], offv = par_s[4], wnorm = par_s[5];

  int wave = tid >> 5, lane = tid & 31;
  int n0 = blockIdx.x * 256 + wave * 32;     // wave handles 32 points
  int n = n0 + lane;

  float u, v, z; rot_view(x, b, n, view, u, v, z);
  float fx = floorf((u - minu) / gs) + 1.0f + offu;
  float fy = floorf((v - minv) / gs) + 1.0f + offv;
  int ix = min(max((int)fx, 0), IMG-1);
  int iy = min(max((int)fy, 0), IMG-1);

  // bicubic upsample value of lr at hi-res pixel (ix,iy); half-pixel centers,
  // out-of-range taps dropped and weights renormalized (jax.image.resize style)
  float G;
  {
    float xi = ((float)ix + 0.5f)*0.25f - 0.5f;
    float yi = ((float)iy + 0.5f)*0.25f - 0.5f;
    float bx = floorf(xi), by = floorf(yi);
    float wx[4], wy[4]; cubic_w(xi - bx, wx); cubic_w(yi - by, wy);
    int ix0 = (int)bx - 1, iy0 = (int)by - 1;
    float sx = 0.f, sy = 0.f;
    float wxx[4], wyy[4]; int txx[4], tyy[4];
    #pragma unroll
    for (int j = 0; j < 4; ++j) {
      int t1 = ix0 + j; bool okx = (t1 >= 0 && t1 < LRW);
      wxx[j] = okx ? wx[j] : 0.f; sx += wxx[j];
      txx[j] = min(max(t1, 0), LRW-1);
      int t2 = iy0 + j; bool oky = (t2 >= 0 && t2 < LRW);
      wyy[j] = oky ? wy[j] : 0.f; sy += wyy[j];
      tyy[j] = min(max(t2, 0), LRW-1);
    }
    float acc = 0.f;
    #pragma unroll
    for (int jy = 0; jy < 4; ++jy) {
      float rowacc = 0.f;
      #pragma unroll
      for (int jx = 0; jx < 4; ++jx)
        rowacc += wxx[jx] * lr_s[txx[jx]*LRW + tyy[jy]];
      acc += wyy[jy] * rowacc;
    }
    G = acc / (sx * sy);
  }
  float tval = G / fmaxf(fabsf(G) * wnorm, 1e-12f);

  // A-matrices (16x4 f32): t in K=0 column => VGPR0 lanes 0-15, all else zero.
  float t_hi = __shfl_down(tval, 16, 32);
  v2f a1; a1.x = (lane < 16) ? tval : 0.f; a1.y = 0.f;
  v2f a2; a2.x = (lane < 16) ? t_hi : 0.f; a2.y = 0.f;

  int lanecol = lane & 15;
  int rowoff  = (lane < 16) ? 0 : 8;
  int colbase = view * CVIEW;
  float* outb = out + (size_t)b * NPTS * CTOT;

  for (int blk = 0; blk < CVIEW/16; ++blk) {
    // B-matrix (4x16 f32): wsum block in K=0 row => VGPR0 lanes 0-15.
    v2f bm; bm.x = (lane < 16) ? wsum_s[blk*16 + lanecol] : 0.f; bm.y = 0.f;
    v8f c1 = {0.f,0.f,0.f,0.f,0.f,0.f,0.f,0.f};
    v8f c2 = {0.f,0.f,0.f,0.f,0.f,0.f,0.f,0.f};
    c1 = __builtin_amdgcn_wmma_f32_16x16x4_f32(false, a1, false, bm, (short)0, c1, false, false);
    c2 = __builtin_amdgcn_wmma_f32_16x16x4_f32(false, a2, false, bm, (short)0, c2, false, false);
    int col = colbase + blk*16 + lanecol;
    #pragma unroll
    for (int r = 0; r < 8; ++r) {
      int m = rowoff + r;                       // D layout: VGPR r = row m, lane = col
      outb[(size_t)(n0 + m)      * CTOT + col] = c1[r];
      outb[(size_t)(n0 + 16 + m) * CTOT + col] = c2[r];
    }
  }
}

// ---------------- host launcher ----------------
extern "C" void kernel_launch(void* const* d_in, const int* in_sizes, int n_in,
                              void* d_out, int out_size, void* d_ws, size_t ws_size,
                              hipStream_t stream) {
  (void)in_sizes; (void)n_in; (void)out_size; (void)ws_size;
  const float* x   = (const float*)d_in[0];   // (4,3,16384) f32
  const float* Wup = (const float*)d_in[1];   // (384,3) f32
  float* out = (float*)d_out;                 // (4,16384,1152) f32
  float* ws  = (float*)d_ws;

  size_t scatter_lds = (size_t)(SPIX + 32) * sizeof(float);   // ~201 KB, < 320 KB WGP LDS
  hipFuncSetAttribute((const void*)scatter_pool_kernel,
                      hipFuncAttributeMaxDynamicSharedMemorySize, (int)scatter_lds);

  prep_w_kernel<<<1, 512, 0, stream>>>(Wup, ws);
  params_kernel<<<12, 256, 0, stream>>>(x, ws);
  scatter_pool_kernel<<<12, 512, scatter_lds, stream>>>(x, ws);
  dim3 egrid(NPTS / 256, 12);
  expand_kernel<<<egrid, 256, 0, stream>>>(x, ws, out);
}